// PatchNet_40458591928354
// MI455X (gfx1250) — compile-verified
//
#include <hip/hip_runtime.h>
#include <cstdint>

typedef __attribute__((ext_vector_type(2))) float v2f;
typedef __attribute__((ext_vector_type(8))) float v8f;

#define NB 32
#define NC 3
#define IH 64
#define IW 64

__device__ __forceinline__ void wait_async0() {
#if __has_builtin(__builtin_amdgcn_s_wait_asynccnt)
  __builtin_amdgcn_s_wait_asynccnt(0);
#else
  asm volatile("s_wait_asynccnt 0" ::: "memory");
#endif
}

// Copy the selected PSZ x PSZ x NC crop through the CDNA5 async
// global->LDS->global path (ASYNCcnt-tracked), 4B per lane per transfer
// (source rows start at arbitrary j0, so 4B is the guaranteed alignment).
template <int PSZ, int OW>
__device__ __forceinline__ void copy_crop(const float* __restrict__ xb,
                                          float* __restrict__ ob,
                                          int n, float* sh) {
  const int i0 = n / OW;
  const int j0 = n - i0 * OW;
  constexpr int total = NC * PSZ * PSZ;
  const int tid = threadIdx.x;
  const uint32_t lds_off = (uint32_t)(size_t)(&sh[tid]);  // low 32b = LDS offset

  for (int base = 0; base < total; base += 256) {
    int k = base + tid;
    if (k >= total) k = total - 1;  // clamped lanes: duplicate same-value store
    const int c = k / (PSZ * PSZ);
    const int r = k - c * (PSZ * PSZ);
    const int i = r / PSZ;
    const int j = r - i * PSZ;

    const float* src = xb + ((size_t)c * IH + (i0 + i)) * IW + (j0 + j);
    asm volatile("global_load_async_to_lds_b32 %0, %1, off"
                 :: "v"(lds_off), "v"(src) : "memory");
    wait_async0();  // async engine has written LDS

    float* dst = ob + k;
    asm volatile("global_store_async_from_lds_b32 %0, %1, off"
                 :: "v"(dst), "v"(lds_off) : "memory");
    wait_async0();  // LDS slot free for next iteration
  }
}

// ---------------------------------------------------------------------------
// Single fused kernel. Grid: 128 blocks = psi(4) x b(32), 256 threads.
//
// Phase 1 (wave 0 only): index[b'] = sum_n idx[b',n]*n for the 16-batch group
// containing b, via V_WMMA_F32_16X16X4_F32 (exact: {0,1} weights, n < 2^24).
//   A (16x4, M=batch, K=chunk): lanes 0-15 hold K=0,1; lanes 16-31 hold K=2,3.
//   B (4x16): position value broadcast over N columns (same K striping).
//   All n values are 1 mod 4 -> branch-free main loop over n-1, one tail WMMA.
// Phase 2 (all 8 waves): async-LDS crop gather.
// ---------------------------------------------------------------------------
__global__ __launch_bounds__(256)
void patch_crop_fused(const float* __restrict__ x,
                      const float* __restrict__ idx0,
                      const float* __restrict__ idx1,
                      const float* __restrict__ idx2,
                      const float* __restrict__ idx3,
                      float* __restrict__ out) {
  __shared__ float sh[256];
  __shared__ int s_n;

  const int psi   = blockIdx.x >> 5;
  const int b     = blockIdx.x & 31;
  const int bbase = b & ~15;

  if (threadIdx.x < 32) {  // wave 0: uniform per-wave branch (s_cbranch_execz)
    const float* w;
    int n;
    if (psi == 0)      { w = idx0; n = 289;  }
    else if (psi == 1) { w = idx1; n = 81;   }
    else if (psi == 2) { w = idx2; n = 1681; }
    else               { w = idx3; n = 1089; }

    const int lane = threadIdx.x;
    const int m    = lane & 15;
    const int kb   = (lane >> 4) << 1;  // 0 for lanes 0-15, 2 for lanes 16-31
    const float* row = w + (size_t)(bbase + m) * n;

    v8f acc = {};
    const int nmain = n - 1;  // multiple of 4: no bounds checks in main loop
    for (int k0 = 0; k0 < nmain; k0 += 4) {
      const int k = k0 + kb;
      v2f a, bm;
      a.x  = row[k];
      a.y  = row[k + 1];
      bm.x = (float)k;
      bm.y = (float)(k + 1);
      acc = __builtin_amdgcn_wmma_f32_16x16x4_f32(
          false, a, false, bm, (short)0, acc, false, false);
    }
    {  // tail chunk: only K = nmain is real; lane ternary -> v_cndmask
      const float last = row[nmain];
      v2f a, bm;
      a.x  = (kb == 0) ? last : 0.0f;
      a.y  = 0.0f;
      bm.x = (float)nmain;
      bm.y = 0.0f;
      acc = __builtin_amdgcn_wmma_f32_16x16x4_f32(
          false, a, false, bm, (short)0, acc, false, false);
    }

    // D layout: VGPR r = M=r (lanes 0-15) / M=8+r (lanes 16-31), N = lane%16.
    const int mloc = b - bbase;  // uniform
    float v;
    switch (mloc & 7) {
      case 0: v = acc[0]; break;
      case 1: v = acc[1]; break;
      case 2: v = acc[2]; break;
      case 3: v = acc[3]; break;
      case 4: v = acc[4]; break;
      case 5: v = acc[5]; break;
      case 6: v = acc[6]; break;
      default: v = acc[7]; break;
    }
    if (lane == ((mloc & 8) ? 16 : 0)) s_n = (int)(v + 0.5f);
  }
  __syncthreads();
  const int n = s_n;

  const float* xb = x + (size_t)b * NC * IH * IW;
  switch (psi) {  // uniform; compile-time strides inside
    case 0: copy_crop<48, 17>(xb, out + 0      + (size_t)b * (NC * 48 * 48), n, sh); break;
    case 1: copy_crop<56,  9>(xb, out + 221184 + (size_t)b * (NC * 56 * 56), n, sh); break;
    case 2: copy_crop<24, 41>(xb, out + 522240 + (size_t)b * (NC * 24 * 24), n, sh); break;
    default: copy_crop<32, 33>(xb, out + 577536 + (size_t)b * (NC * 32 * 32), n, sh); break;
  }
}

// ---------------------------------------------------------------------------
extern "C" void kernel_launch(void* const* d_in, const int* in_sizes, int n_in,
                              void* d_out, int out_size, void* d_ws, size_t ws_size,
                              hipStream_t stream) {
  const float* x    = (const float*)d_in[0];
  const float* idx0 = (const float*)d_in[1];
  const float* idx1 = (const float*)d_in[2];
  const float* idx2 = (const float*)d_in[3];
  const float* idx3 = (const float*)d_in[4];
  float* out = (float*)d_out;
  (void)d_ws; (void)ws_size; (void)in_sizes; (void)n_in; (void)out_size;

  patch_crop_fused<<<dim3(4 * NB), dim3(256), 0, stream>>>(x, idx0, idx1, idx2,
                                                           idx3, out);
}